// MultiBoxLoss_38233798869291
// MI455X (gfx1250) — compile-verified
//
#include <hip/hip_runtime.h>
#include <hip/hip_bf16.h>

typedef __attribute__((ext_vector_type(2))) float v2f;
typedef __attribute__((ext_vector_type(8))) float v8f;

#define NOBJ 32

__device__ inline float sl1f(float x) {
    float a = fabsf(x);
    return a < 1.f ? 0.5f * x * x : a - 0.5f;
}

__device__ inline float iou_xyxy(float ax0, float ay0, float ax1, float ay1,
                                 float bx0, float by0, float bx1, float by1) {
    float lx = fmaxf(ax0, bx0), ly = fmaxf(ay0, by0);
    float rx = fminf(ax1, bx1), ry = fminf(ay1, by1);
    float w = fmaxf(rx - lx, 0.f), h = fmaxf(ry - ly, 0.f);
    float inter = w * h;
    float aa = (ax1 - ax0) * (ay1 - ay0);
    float ab = (bx1 - bx0) * (by1 - by0);
    return inter / (aa + ab - inter);
}

// Block-wide (256 threads) sum of per-thread partials on the FP32 WMMA pipe.
// A(16x4) = data, B(4x16) = ones  ->  D[m][n] = rowsum_m(A) for every n.
// Chained over 4 tiles (256 values). Per-lane sum of the 8 D VGPRs gives
// sum of rowsums M=0..7 (lanes 0-15) / M=8..15 (lanes 16-31); adding the
// lane^16 partner yields the grand total. Exact FP32, RNE.
// Executed by wave 0 only, with full EXEC (threads 0..31 all active).
// Result valid on threads 0..31.
__device__ float block_reduce_wmma256(float v, float* red) {
    red[threadIdx.x] = v;
    __syncthreads();
    float out = 0.f;
    if (threadIdx.x < 32) {
        v8f c = {};
        v2f ones;
        ones.x = 1.f; ones.y = 1.f;
#pragma unroll
        for (int i = 0; i < 4; ++i) {
            v2f a;
            a.x = red[i * 64 + threadIdx.x * 2 + 0];
            a.y = red[i * 64 + threadIdx.x * 2 + 1];
            // (neg_a, A, neg_b, B, c_mod, C, reuse_a, reuse_b)
            c = __builtin_amdgcn_wmma_f32_16x16x4_f32(
                    false, a, false, ones, (short)0, c, false, false);
        }
        float s = c[0] + c[1] + c[2] + c[3] + c[4] + c[5] + c[6] + c[7];
        s += __shfl_xor(s, 16, 32);
        out = s;
    }
    __syncthreads();
    return out;
}

// One block per image. Matching + forced-prior scatter + fused loc loss.
__global__ __launch_bounds__(256) void k_match(
        const float* __restrict__ loc_pred,
        const float* __restrict__ gt_boxes,
        const int*   __restrict__ gt_labels,
        const float* __restrict__ dbox,
        int*   __restrict__ true_labels,
        int*   __restrict__ n_pos_arr,
        float* __restrict__ f_acc,
        int*   __restrict__ i_acc,
        int P) {
    __shared__ float s_gx0[NOBJ], s_gy0[NOBJ], s_gx1[NOBJ], s_gy1[NOBJ];
    __shared__ int s_lab[NOBJ];
    __shared__ unsigned long long s_best[NOBJ];
    __shared__ int s_forced[NOBJ];
    __shared__ int s_npos;
    __shared__ float s_red[256];

    const int b = blockIdx.x;
    const int tid = threadIdx.x;
    if (tid < NOBJ) {
        const float* g = gt_boxes + ((size_t)b * NOBJ + tid) * 4;
        s_gx0[tid] = g[0]; s_gy0[tid] = g[1]; s_gx1[tid] = g[2]; s_gy1[tid] = g[3];
        s_lab[tid] = gt_labels[b * NOBJ + tid];
        s_best[tid] = 0ull;
    }
    if (tid == 0) s_npos = 0;
    __syncthreads();

    // Phase 1: per-object best prior (value bits << 32 | ~index; ties -> lowest index)
    float bv[NOBJ];
    int bp[NOBJ];
#pragma unroll
    for (int o = 0; o < NOBJ; ++o) { bv[o] = -1.f; bp[o] = 0; }
    for (int p = tid; p < P; p += 256) {
        const float* d = dbox + (size_t)p * 4;
        float cx = d[0], cy = d[1], w = d[2], h = d[3];
        float x0 = cx - 0.5f * w, y0 = cy - 0.5f * h;
        float x1 = cx + 0.5f * w, y1 = cy + 0.5f * h;
#pragma unroll
        for (int o = 0; o < NOBJ; ++o) {
            float v = iou_xyxy(s_gx0[o], s_gy0[o], s_gx1[o], s_gy1[o], x0, y0, x1, y1);
            if (v > bv[o]) { bv[o] = v; bp[o] = p; }
        }
    }
#pragma unroll
    for (int o = 0; o < NOBJ; ++o) {
        unsigned long long pk =
            ((unsigned long long)__float_as_uint(fmaxf(bv[o], 0.f)) << 32) |
            (unsigned long long)(~(unsigned)bp[o]);
        atomicMax(&s_best[o], pk);
    }
    __syncthreads();
    if (tid == 0) {
        for (int o = 0; o < NOBJ; ++o)
            s_forced[o] = (int)(~(unsigned)(s_best[o] & 0xFFFFFFFFull));
    }
    __syncthreads();

    // Phase 2: per-prior best object (recompute; compute is free), force, label,
    // fused smooth-L1 loc loss on positives.
    float loc_part = 0.f;
    int cnt = 0;
    for (int p = tid; p < P; p += 256) {
        const float* d = dbox + (size_t)p * 4;
        float cx = d[0], cy = d[1], w = d[2], h = d[3];
        float x0 = cx - 0.5f * w, y0 = cy - 0.5f * h;
        float x1 = cx + 0.5f * w, y1 = cy + 0.5f * h;
        float ov = -1.f;
        int obj = 0;
#pragma unroll
        for (int o = 0; o < NOBJ; ++o) {
            float v = iou_xyxy(s_gx0[o], s_gy0[o], s_gx1[o], s_gy1[o], x0, y0, x1, y1);
            if (v > ov) { ov = v; obj = o; }
        }
#pragma unroll
        for (int o = 0; o < NOBJ; ++o)
            if (s_forced[o] == p) { obj = o; ov = 1.f; }  // ascending: last wins
        int lab = (ov < 0.5f) ? 0 : s_lab[obj];
        true_labels[(size_t)b * P + p] = lab;
        if (lab > 0) {
            ++cnt;
            float gx0 = s_gx0[obj], gy0 = s_gy0[obj], gx1 = s_gx1[obj], gy1 = s_gy1[obj];
            float gcx = 0.5f * (gx0 + gx1), gcy = 0.5f * (gy0 + gy1);
            float gw = gx1 - gx0, gh = gy1 - gy0;
            float t0 = (gcx - cx) * 10.f / w;
            float t1 = (gcy - cy) * 10.f / h;
            float t2 = __logf(gw / w) * 5.f;
            float t3 = __logf(gh / h) * 5.f;
            const float* lp = loc_pred + ((size_t)b * P + p) * 4;
            loc_part += sl1f(lp[0] - t0) + sl1f(lp[1] - t1) +
                        sl1f(lp[2] - t2) + sl1f(lp[3] - t3);
        }
    }
    atomicAdd(&s_npos, cnt);
    float tot = block_reduce_wmma256(loc_part, s_red);
    if (tid == 0) {
        atomicAdd(&f_acc[0], tot);     // loc loss numerator
        n_pos_arr[b] = s_npos;
        atomicAdd(&i_acc[0], s_npos);  // n_pos_total
    }
}

// One wave32 per (b,p) row: log-softmax over C=81, CE; single coalesced read
// of cls_pred (the 181 MB / 7.8 us HBM roofline term).
__global__ __launch_bounds__(256) void k_ce(
        const float* __restrict__ cls_pred,
        const int*   __restrict__ true_labels,
        float* __restrict__ ce_neg,
        float* __restrict__ f_acc,
        long long nrows, int C) {
    __shared__ float s_red[256];
    const int lane = threadIdx.x & 31;
    const int wid = threadIdx.x >> 5;
    long long row = (long long)blockIdx.x * 8 + wid;
    float pos_part = 0.f;
    if (row < nrows) {
        const float* x = cls_pred + (size_t)row * C;
        float x0 = x[lane];
        float x1 = x[lane + 32];
        bool has2 = (lane + 64) < C;
        float x2 = has2 ? x[lane + 64] : -3.4e38f;
        float m = fmaxf(fmaxf(x0, x1), x2);
#pragma unroll
        for (int d = 16; d > 0; d >>= 1) m = fmaxf(m, __shfl_xor(m, d, 32));
        float e = __expf(x0 - m) + __expf(x1 - m) + (has2 ? __expf(x2 - m) : 0.f);
#pragma unroll
        for (int d = 16; d > 0; d >>= 1) e += __shfl_xor(e, d, 32);
        float lse = __logf(e) + m;
        int lab = true_labels[row];
        float ce = lse - x[lab];
        if (lane == 0) {
            if (lab > 0) { pos_part = ce; ce_neg[row] = 0.f; }
            else         { ce_neg[row] = ce; }
        }
    }
    float tot = block_reduce_wmma256(pos_part, s_red);
    if (threadIdx.x == 0) atomicAdd(&f_acc[1], tot);  // conf_pos
}

// One block per image: exact top-K sum (K = 3*n_pos) of ce_neg via 4-pass
// radix select on float bits (ce >= 0 => u32-order == float-order).
__global__ __launch_bounds__(256) void k_hardneg(
        const float* __restrict__ ce_neg,
        const int*   __restrict__ n_pos_arr,
        float* __restrict__ f_acc,
        int P) {
    __shared__ unsigned s_hist[256];
    __shared__ unsigned s_prefix;
    __shared__ int s_krem;
    __shared__ int s_cnt;
    __shared__ float s_red[256];
    const int b = blockIdx.x;
    const int tid = threadIdx.x;
    const float* v = ce_neg + (size_t)b * P;
    int K = 3 * n_pos_arr[b];
    if (K > P) K = P;
    if (K <= 0) return;  // uniform across block
    if (tid == 0) { s_prefix = 0u; s_krem = K; s_cnt = 0; }
    __syncthreads();
    for (int pass = 0; pass < 4; ++pass) {
        const int shift = 24 - pass * 8;
        s_hist[tid] = 0u;
        __syncthreads();
        unsigned hi_mask = (pass == 0) ? 0u : (0xFFFFFFFFu << (32 - 8 * pass));
        unsigned pref = s_prefix;
        for (int i = tid; i < P; i += 256) {
            unsigned bits = __float_as_uint(v[i]);
            if ((bits & hi_mask) == pref)
                atomicAdd(&s_hist[(bits >> shift) & 255], 1u);
        }
        __syncthreads();
        if (tid == 0) {
            int krem = s_krem;
            unsigned digit = 0;
            for (int d = 255; d >= 0; --d) {
                int c = (int)s_hist[d];
                if (krem <= c) { digit = (unsigned)d; break; }
                krem -= c;
            }
            s_krem = krem;
            s_prefix = pref | (digit << shift);
        }
        __syncthreads();
    }
    const unsigned T = s_prefix;          // exact K-th largest value (bits)
    const float Tf = __uint_as_float(T);
    float sum_part = 0.f;
    int cnt_part = 0;
    for (int i = tid; i < P; i += 256) {
        float val = v[i];
        if (__float_as_uint(val) > T) { sum_part += val; ++cnt_part; }
    }
    atomicAdd(&s_cnt, cnt_part);
    float s = block_reduce_wmma256(sum_part, s_red);
    if (tid == 0) {
        float hard = s + (float)(K - s_cnt) * Tf;  // ties contribute T each
        atomicAdd(&f_acc[2], hard);                // conf_hard
    }
}

__global__ void k_zero(float* f_acc, int* i_acc) {
    if (threadIdx.x < 4) f_acc[threadIdx.x] = 0.f;
    if (threadIdx.x == 4) i_acc[0] = 0;
}

__global__ void k_final(const float* __restrict__ f_acc,
                        const int* __restrict__ i_acc,
                        float* __restrict__ out) {
    float npt = (float)i_acc[0];
    out[0] = (f_acc[1] + f_acc[2]) / npt + f_acc[0] / (npt * 4.f);
}

extern "C" void kernel_launch(void* const* d_in, const int* in_sizes, int n_in,
                              void* d_out, int out_size, void* d_ws, size_t ws_size,
                              hipStream_t stream) {
    const float* loc_pred  = (const float*)d_in[0];
    const float* cls_pred  = (const float*)d_in[1];
    const float* gt_boxes  = (const float*)d_in[2];
    const int*   gt_labels = (const int*)d_in[3];
    const float* dbox      = (const float*)d_in[4];

    const int P = in_sizes[4] / 4;
    const long long BP = (long long)in_sizes[0] / 4;  // B*P
    const int Bv = (int)(BP / P);
    const int C = (int)((long long)in_sizes[1] / BP);

    char* ws = (char*)d_ws;
    float* f_acc = (float*)ws;          // [0]=loc_sum [1]=conf_pos [2]=conf_hard
    int* i_acc = (int*)(ws + 16);       // [0]=n_pos_total
    int* n_pos = (int*)(ws + 32);       // [B]
    size_t off = 512;
    int* true_labels = (int*)(ws + off);
    off += (size_t)BP * sizeof(int);
    off = (off + 255) & ~(size_t)255;
    float* ce_neg = (float*)(ws + off);

    k_zero<<<1, 64, 0, stream>>>(f_acc, i_acc);
    k_match<<<Bv, 256, 0, stream>>>(loc_pred, gt_boxes, gt_labels, dbox,
                                    true_labels, n_pos, f_acc, i_acc, P);
    long long nblocks = (BP + 7) / 8;
    k_ce<<<dim3((unsigned)nblocks), 256, 0, stream>>>(cls_pred, true_labels,
                                                      ce_neg, f_acc, BP, C);
    k_hardneg<<<Bv, 256, 0, stream>>>(ce_neg, n_pos, f_acc, P);
    k_final<<<1, 1, 0, stream>>>(f_acc, i_acc, (float*)d_out);
}